// IrrepMultiHeadAttention_10857677324507
// MI455X (gfx1250) — compile-verified
//
#include <hip/hip_runtime.h>

// Problem constants (match reference)
constexpr int BB = 2;
constexpr int NN = 128;
constexpr int KK = 32;
constexpr float KAPPA_INV = 1.0f;     // kappa = 1.0
constexpr float EPSR     = 1e-8f;     // cholesky regularization
constexpr float EPS_SOFT = 1e-8f;     // softmax renorm epsilon

#define LDT 33                         // padded LDS row stride (odd -> conflict-free)

typedef __attribute__((ext_vector_type(2))) float    f32x2;
typedef __attribute__((ext_vector_type(8))) float    f32x8;
typedef __attribute__((ext_vector_type(4))) unsigned u32x4;
typedef __attribute__((ext_vector_type(8))) unsigned u32x8;

// ---------------------------------------------------------------------------
// LDS byte-offset of a __shared__ object (generic -> addrspace(3) cast strips
// the aperture; ptrtoint of an AS3 pointer is the raw LDS offset).
// ---------------------------------------------------------------------------
__device__ __forceinline__ unsigned lds_offset_of(const void* p)
{
    return (unsigned)(unsigned long long)
           (__attribute__((address_space(3))) const void*)p;
}

// ---------------------------------------------------------------------------
// Tensor Data Mover: DMA one dense 32x32 f32 tile (row-major, stride 32
// elements) from global memory into LDS at byte offset lds_off.
// D# layout per CDNA5 ISA:
//   group0: [1:0]=count=1, [63:32]=lds_addr, [120:64]=global_addr,
//           [127:126]=type=2
//   group1: data_size=2 (4B), tensor_dim0=32, tensor_dim1=32, tile_dim0=32,
//           tile_dim1=32, tensor_dim0_stride=32, no pad / no multicast.
// Issued by ONE wave; tracked with TENSORcnt.
// ---------------------------------------------------------------------------
__device__ __forceinline__ void tdm_load_tile32(const float* gsrc, unsigned lds_off)
{
    const unsigned long long ga = (unsigned long long)gsrc;
    u32x4 g0;
    g0[0] = 1u;                                             // count = 1 (valid)
    g0[1] = lds_off;                                        // lds_addr
    g0[2] = (unsigned)ga;                                   // global_addr[31:0]
    g0[3] = ((unsigned)(ga >> 32) & 0x01FFFFFFu)            // global_addr[56:32]
          | 0x80000000u;                                    // type = 2 ("image")
    u32x8 g1;
    g1[0] = 0x00020000u;      // workgroup_mask=0, data_size=2 (4 bytes)
    g1[1] = (32u << 16);      // tensor_dim0[15:0] = 32
    g1[2] = (32u << 16);      // tensor_dim1[15:0] = 32
    g1[3] = (32u << 16);      // tile_dim0 = 32
    g1[4] = 32u;              // tile_dim1 = 32, tile_dim2 = 0
    g1[5] = 32u;              // tensor_dim0_stride = 32
    g1[6] = 0u;
    g1[7] = 0u;
    asm volatile("tensor_load_to_lds %0, %1" :: "s"(g0), "s"(g1) : "memory");
}

// ---------------------------------------------------------------------------
// 32x32x32 f32 GEMM in LDS using V_WMMA_F32_16X16X4_F32.
// 128 threads = 4 waves; wave w owns output tile (w>>1, w&1) of 16x16.
// A is MxK row-major (stride LDT), B is KxN row-major with stride ldb, or
// NxK if TRANSB (C = A * B^T). C row-major stride LDT. Caller syncs around.
// Fragment layouts per CDNA5 ISA 7.12.2 (f32):
//   A 16x4 : lane l -> m = l&15, k = 2*(l>>4) + {0,1}
//   B 4x16 : lane l -> n = l&15, k = 2*(l>>4) + {0,1}
//   C 16x16: lane l -> n = l&15, vgpr v -> m = v + 8*(l>>4)
// ---------------------------------------------------------------------------
template <bool TRANSB>
__device__ __forceinline__ void gemm32x32(const float* __restrict__ As,
                                          const float* __restrict__ Bs,
                                          float* __restrict__ Cs,
                                          int ldb)
{
    const int tid  = threadIdx.x;
    const int wave = tid >> 5;
    const int lane = tid & 31;
    const int m0   = (wave >> 1) << 4;
    const int n0   = (wave & 1) << 4;
    const int hl   = lane >> 4;        // 0 or 1
    const int l16  = lane & 15;

    f32x8 acc = {};
#pragma unroll
    for (int kk = 0; kk < 32; kk += 4) {
        const int kb = kk + 2 * hl;
        f32x2 a, b;
        a[0] = As[(m0 + l16) * LDT + kb + 0];
        a[1] = As[(m0 + l16) * LDT + kb + 1];
        if (!TRANSB) {
            b[0] = Bs[(kb + 0) * ldb + n0 + l16];
            b[1] = Bs[(kb + 1) * ldb + n0 + l16];
        } else {
            b[0] = Bs[(n0 + l16) * ldb + kb + 0];
            b[1] = Bs[(n0 + l16) * ldb + kb + 1];
        }
        acc = __builtin_amdgcn_wmma_f32_16x16x4_f32(
                  false, a, false, b, (short)0, acc, false, false);
    }
#pragma unroll
    for (int v = 0; v < 8; ++v) {
        const int m = v + 8 * hl;
        Cs[(m0 + m) * LDT + n0 + l16] = acc[v];
    }
}

// ---------------------------------------------------------------------------
// exp(64*Xs) via order-8 Horner Taylor + 6 repeated squarings.
// Xs must hold the PRE-SCALED matrix (X / 2^6). Result in Ts. Us is scratch.
// ---------------------------------------------------------------------------
__device__ void expm32(float* Xs, float* Ts, float* Us)
{
    const int tid = threadIdx.x;
    for (int idx = tid; idx < KK * KK; idx += 128) {
        const int r = idx >> 5, c = idx & 31;
        Ts[r * LDT + c] = (r == c) ? 1.0f : 0.0f;
    }
    __syncthreads();
    for (int k = 8; k >= 1; --k) {
        gemm32x32<false>(Xs, Ts, Us, LDT);      // Us = X' * T
        __syncthreads();
        const float inv = 1.0f / (float)k;
        for (int idx = tid; idx < KK * KK; idx += 128) {
            const int r = idx >> 5, c = idx & 31;
            Ts[r * LDT + c] = ((r == c) ? 1.0f : 0.0f) + Us[r * LDT + c] * inv;
        }
        __syncthreads();
    }
    for (int s = 0; s < 6; ++s) {               // squaring phase
        gemm32x32<false>(Ts, Ts, Us, LDT);
        __syncthreads();
        for (int idx = tid; idx < KK * KK; idx += 128) {
            const int r = idx >> 5, c = idx & 31;
            Ts[r * LDT + c] = Us[r * LDT + c];
        }
        __syncthreads();
    }
}

// In-place lower Cholesky of a 32x32 SPD matrix in LDS (stride LDT).
__device__ void cholesky32(float* A)
{
    const int tid = threadIdx.x;
    for (int k = 0; k < KK; ++k) {
        if (tid == 0) A[k * LDT + k] = sqrtf(fmaxf(A[k * LDT + k], 1e-30f));
        __syncthreads();
        const float dk = A[k * LDT + k];
        if (tid > k && tid < KK) A[tid * LDT + k] /= dk;
        __syncthreads();
        for (int idx = tid; idx < KK * KK; idx += 128) {
            const int r = idx >> 5, c = idx & 31;
            if (r > k && c > k && c <= r)
                A[r * LDT + c] -= A[r * LDT + k] * A[c * LDT + k];
        }
        __syncthreads();
    }
}

// ---------------------------------------------------------------------------
// Kernel 1: per-agent precompute. grid = B*N blocks, 128 threads.
// ---------------------------------------------------------------------------
__global__ __launch_bounds__(128)
void irrep_precompute_kernel(const float* __restrict__ mu_q,
                             const float* __restrict__ sigma_q,
                             const float* __restrict__ phi,
                             const float* __restrict__ gens,
                             float* __restrict__ E_g,
                             float* __restrict__ S_g,
                             float* __restrict__ v_g,
                             float* __restrict__ L1_g,
                             float* __restrict__ ld1_g)
{
    __shared__ float Xs[KK * LDT], Ts[KK * LDT], Us[KK * LDT], Sig[KK * LDT];
    __shared__ float mus[KK];

    const int agent = blockIdx.x;            // b*N + n
    const int tid   = threadIdx.x;

    const float p0 = phi[agent * 3 + 0];
    const float p1 = phi[agent * 3 + 1];
    const float p2 = phi[agent * 3 + 2];
    for (int idx = tid; idx < KK * KK; idx += 128) {
        const float x = p0 * gens[idx] + p1 * gens[KK * KK + idx]
                      + p2 * gens[2 * KK * KK + idx];
        Xs[(idx >> 5) * LDT + (idx & 31)] = x * 0.015625f;   // X / 2^6
    }
    if (tid < KK) mus[tid] = mu_q[agent * KK + tid];
    __syncthreads();

    // E = exp(X)
    expm32(Xs, Ts, Us);
    for (int idx = tid; idx < KK * KK; idx += 128)
        E_g[agent * KK * KK + idx] = Ts[(idx >> 5) * LDT + (idx & 31)];
    __syncthreads();

    // Em = exp(-X)
    for (int idx = tid; idx < KK * KK; idx += 128)
        Xs[(idx >> 5) * LDT + (idx & 31)] = -Xs[(idx >> 5) * LDT + (idx & 31)];
    __syncthreads();
    expm32(Xs, Ts, Us);                       // Ts = Em

    for (int idx = tid; idx < KK * KK; idx += 128)
        Sig[(idx >> 5) * LDT + (idx & 31)] = sigma_q[agent * KK * KK + idx];
    __syncthreads();

    // v = Em * mu
    if (tid < KK) {
        float acc = 0.0f;
        for (int c = 0; c < KK; ++c) acc += Ts[tid * LDT + c] * mus[c];
        v_g[agent * KK + tid] = acc;
    }

    // S = Em * Sig * Em^T  (two WMMA GEMMs)
    gemm32x32<false>(Ts, Sig, Us, LDT);       // Us = Em * Sig
    __syncthreads();
    gemm32x32<true>(Us, Ts, Xs, LDT);         // Xs = Us * Em^T
    __syncthreads();
    for (int idx = tid; idx < KK * KK; idx += 128)
        S_g[agent * KK * KK + idx] = Xs[(idx >> 5) * LDT + (idx & 31)];

    // L1 = chol(Sig + eps I), logdet1
    if (tid < KK) Sig[tid * LDT + tid] += EPSR;
    __syncthreads();
    cholesky32(Sig);
    for (int idx = tid; idx < KK * KK; idx += 128) {
        const int r = idx >> 5, c = idx & 31;
        L1_g[agent * KK * KK + idx] = (c <= r) ? Sig[r * LDT + c] : 0.0f;
    }
    if (tid == 0) {
        float s = 0.0f;
        for (int k = 0; k < KK; ++k) s += __logf(Sig[k * LDT + k]);
        ld1_g[agent] = 2.0f * s;
    }
}

// ---------------------------------------------------------------------------
// Kernel 2: per-row attention. grid = B*N blocks (one per (b,i)), 128 threads.
// S_j tiles are streamed into LDS by the Tensor Data Mover (double-buffered,
// dense stride-32 tiles), overlapping DMA of tile j+1 with compute on tile j.
// ---------------------------------------------------------------------------
__global__ __launch_bounds__(128)
void irrep_attend_kernel(const float* __restrict__ mu_q,
                         const float* __restrict__ E_g,
                         const float* __restrict__ S_g,
                         const float* __restrict__ v_g,
                         const float* __restrict__ L1_g,
                         const float* __restrict__ ld1_g,
                         float* __restrict__ out)
{
    __shared__ float Ei[KK * LDT], L1i[KK * LDT];
    __shared__ float T1[KK * LDT], T2[KK * LDT];
    __shared__ alignas(16) float Sjd[2][KK * KK];   // dense TDM tiles (stride 32)
    __shared__ float mtall[NN][KK];                 // transported means for all j
    __shared__ float mui[KK], vj[KK];
    __shared__ float klArr[NN];
    __shared__ float red[KK + 1];
    __shared__ float sLd2, sMax, sSum;

    const int bi  = blockIdx.x;               // b*N + i
    const int b   = bi >> 7;                  // N = 128
    const int tid = threadIdx.x;

    for (int idx = tid; idx < KK * KK; idx += 128) {
        const int r = idx >> 5, c = idx & 31;
        Ei[r * LDT + c]  = E_g[bi * KK * KK + idx];
        L1i[r * LDT + c] = L1_g[bi * KK * KK + idx];
    }
    if (tid < KK) mui[tid] = mu_q[bi * KK + tid];
    const float logdet1 = ld1_g[bi];

    const unsigned sj_off0 = lds_offset_of(&Sjd[0][0]);
    const unsigned sj_off1 = lds_offset_of(&Sjd[1][0]);

    // Kick off the first TDM tile (wave 0 only; TENSORcnt tracks it).
    if (tid < 32) {
        tdm_load_tile32(&S_g[(size_t)(b << 7) * KK * KK], sj_off0);
    }
    __syncthreads();

    for (int j = 0; j < NN; ++j) {
        const int bj = (b << 7) + j;

        // Wait for this iteration's tile, publish to the workgroup.
        if (tid < 32) __builtin_amdgcn_s_wait_tensorcnt(0);
        __syncthreads();

        if (tid < KK) vj[tid] = v_g[bj * KK + tid];
        if (j + 1 < NN) {
            if (tid < 32) {
                // DMA next tile into the other buffer while we compute.
                tdm_load_tile32(&S_g[(size_t)(bj + 1) * KK * KK],
                                ((j + 1) & 1) ? sj_off1 : sj_off0);
            }
            if (tid == 33)                    // prefetch next v_j (global_prefetch_b8)
                __builtin_prefetch(&v_g[(bj + 1) * KK], 0, 1);
        }
        const float* Sjcur = Sjd[j & 1];

        // T2 = E_i * S_j * E_i^T  (two WMMA GEMMs)
        gemm32x32<false>(Ei, Sjcur, T1, KK);  // dense B, stride 32
        __syncthreads();
        gemm32x32<true>(T1, Ei, T2, LDT);
        __syncthreads();

        // symmetrize + eps*I
        for (int idx = tid; idx < KK * KK; idx += 128) {
            const int r = idx >> 5, c = idx & 31;
            if (c <= r) {
                const float a = T2[r * LDT + c];
                const float d = T2[c * LDT + r];
                float m = 0.5f * (a + d);
                if (r == c) m += EPSR;
                T2[r * LDT + c] = m;
                if (c < r) T2[c * LDT + r] = m;
            }
        }
        __syncthreads();

        // transported mean mu_t(i,j) = E_i * v_j
        if (tid < KK) {
            float acc = 0.0f;
            for (int c = 0; c < KK; ++c) acc += Ei[tid * LDT + c] * vj[c];
            mtall[j][tid] = acc;
        }
        __syncthreads();

        cholesky32(T2);                       // T2 := L2 (lower)

        if (tid == 0) {
            float s = 0.0f;
            for (int k = 0; k < KK; ++k) s += __logf(T2[k * LDT + k]);
            sLd2 = 2.0f * s;
        }

        // Forward substitution: columns 0..31 -> trace term, column 32 -> quad
        if (tid < KK + 1) {
            float y[KK];
            float sumsq = 0.0f;
            for (int r = 0; r < KK; ++r) {
                float acc = (tid < KK) ? L1i[r * LDT + tid]
                                       : (mtall[j][r] - mui[r]);
                for (int t = 0; t < r; ++t) acc -= T2[r * LDT + t] * y[t];
                const float yr = acc / T2[r * LDT + r];
                y[r] = yr;
                sumsq += yr * yr;
            }
            red[tid] = sumsq;
        }
        __syncthreads();
        if (tid == 0) {
            float tr = 0.0f;
            for (int c = 0; c < KK; ++c) tr += red[c];
            const float quad = red[KK];
            const float kl = 0.5f * (tr + quad - (float)KK + sLd2 - logdet1);
            klArr[j] = fmaxf(kl, 0.0f);
        }
        __syncthreads();
    }

    // softmax over j of (-kl/kappa), epsilon-renormalize
    if (tid == 0) {
        float mx = -klArr[0] * KAPPA_INV;
        for (int t = 1; t < NN; ++t) mx = fmaxf(mx, -klArr[t] * KAPPA_INV);
        sMax = mx;
    }
    __syncthreads();
    const float e = __expf(-klArr[tid] * KAPPA_INV - sMax);
    __syncthreads();
    klArr[tid] = e;
    __syncthreads();
    if (tid == 0) {
        float s = 0.0f;
        for (int t = 0; t < NN; ++t) s += klArr[t];
        sSum = s;
    }
    __syncthreads();
    const float beta = (klArr[tid] / sSum + EPS_SOFT)
                     / (1.0f + (float)NN * EPS_SOFT);
    __syncthreads();
    klArr[tid] = beta;
    __syncthreads();

    // mu_agg_i = sum_j beta_j * mu_t(i,j)
    if (tid < KK) {
        float acc = 0.0f;
        for (int jj = 0; jj < NN; ++jj) acc += klArr[jj] * mtall[jj][tid];
        out[bi * KK + tid] = acc;
    }
}

// ---------------------------------------------------------------------------
extern "C" void kernel_launch(void* const* d_in, const int* in_sizes, int n_in,
                              void* d_out, int out_size, void* d_ws, size_t ws_size,
                              hipStream_t stream)
{
    (void)in_sizes; (void)n_in; (void)out_size; (void)ws_size;

    const float* mu_q    = (const float*)d_in[0];   // (B,N,K)
    const float* sigma_q = (const float*)d_in[1];   // (B,N,K,K)
    const float* phi     = (const float*)d_in[2];   // (B,N,3)
    const float* gens    = (const float*)d_in[3];   // (3,K,K)
    float* out = (float*)d_out;                     // (B,N,K)

    // workspace layout (floats)
    float* E_g  = (float*)d_ws;                     // B*N*K*K
    float* S_g  = E_g  + BB * NN * KK * KK;         // B*N*K*K
    float* L1_g = S_g  + BB * NN * KK * KK;         // B*N*K*K
    float* v_g  = L1_g + BB * NN * KK * KK;         // B*N*K
    float* ld1  = v_g  + BB * NN * KK;              // B*N

    dim3 grid(BB * NN);
    dim3 block(128);
    irrep_precompute_kernel<<<grid, block, 0, stream>>>(
        mu_q, sigma_q, phi, gens, E_g, S_g, v_g, L1_g, ld1);
    irrep_attend_kernel<<<grid, block, 0, stream>>>(
        mu_q, E_g, S_g, v_g, L1_g, ld1, out);
}